// WCTA_28810640622200
// MI455X (gfx1250) — compile-verified
//
#include <hip/hip_runtime.h>
#include <hip/hip_bf16.h>

#define Bn 16
#define Cc 64
#define HWn 65536
#define SLICES 32
#define ROWS_PER_SLICE (HWn / SLICES) /* 2048 */
#define CHUNK 64
#define XPAD 68 /* padded LDS row stride (floats); 68*4=272 B, 16B-aligned rows */

typedef float v2f __attribute__((ext_vector_type(2)));
typedef float v8f __attribute__((ext_vector_type(8)));
typedef int   v4i __attribute__((__vector_size__(4 * sizeof(int))));

// ---- gfx1250 async LDS data path (guarded; float4 sync fallback) ----
#if defined(__has_builtin)
#if __has_builtin(__builtin_amdgcn_global_load_async_to_lds_b128)
#define HAVE_ASYNC_LOAD 1
#endif
#if __has_builtin(__builtin_amdgcn_global_store_async_from_lds_b128)
#define HAVE_ASYNC_STORE 1
#endif
#endif

#if defined(HAVE_ASYNC_LOAD) || defined(HAVE_ASYNC_STORE)
#if __has_builtin(__builtin_amdgcn_s_wait_asynccnt)
#define WAIT_ASYNC0() __builtin_amdgcn_s_wait_asynccnt(0)
#else
#define WAIT_ASYNC0() asm volatile("s_wait_asynccnt 0x0" ::: "memory")
#endif
#else
#define WAIT_ASYNC0() do {} while (0)
#endif

// global (AS1) / LDS (AS3) int4-vector pointers, matching the builtin prototype
#define GPTR4(p) ((__attribute__((address_space(1))) v4i*)(p))
#define LPTR4(p) ((__attribute__((address_space(3))) v4i*)(p))

__device__ __forceinline__ void ldsLoad16(const float* gsrc, float* ldst) {
#ifdef HAVE_ASYNC_LOAD
  __builtin_amdgcn_global_load_async_to_lds_b128(GPTR4(gsrc), LPTR4(ldst), 0, 0);
#else
  *(float4*)ldst = *(const float4*)gsrc;
#endif
}

__device__ __forceinline__ void ldsStore16(float* gdst, const float* lsrc) {
#ifdef HAVE_ASYNC_STORE
  __builtin_amdgcn_global_store_async_from_lds_b128(GPTR4(gdst), LPTR4(lsrc), 0, 0);
#else
  *(float4*)gdst = *(const float4*)lsrc;
#endif
}

// ---- workspace layout (float offsets) ----
#define MEAN_OFF   0                                  // B*C
#define PSUM_OFF   (MEAN_OFF + Bn*Cc)                 // B*S*C
#define PGRAM_OFF  (PSUM_OFF + Bn*SLICES*Cc)          // B*S*C*C = 2097152
#define COV_OFF    (PGRAM_OFF + Bn*SLICES*Cc*Cc)
#define YSQ_OFF    (COV_OFF + Bn*Cc*Cc)               // cov^{+1/2}
#define ZIN_OFF    (YSQ_OFF + Bn*Cc*Cc)               // cov^{-1/2}
#define MT_OFF     (ZIN_OFF + Bn*Cc*Cc)
#define BIAS_OFF   (MT_OFF + Bn*Cc*Cc)

// ============================================================================
// Kernel 1: partial Gram (X^T X over an HW slice) + partial channel sums.
// Double-buffered async staging of 64x64 f32 tiles; per-wave two 16x16 WMMA
// tiles sharing the A fragment (V_WMMA_F32_16X16X4_F32).
// ============================================================================
__global__ __launch_bounds__(256) void wct_gram_partial(const float* __restrict__ x,
                                                        float* __restrict__ ws) {
  __shared__ float tile[2][CHUNK * XPAD];
  __shared__ float red[256];
  const int b = blockIdx.x / SLICES;
  const int s = blockIdx.x % SLICES;
  const int tid  = threadIdx.x;
  const int lane = tid & 31;
  const int wave = tid >> 5;
  const int m  = lane & 15;
  const int kh = lane >> 4;
  const int ci0  = (wave >> 1) * 16;
  const int cj0a = (wave & 1) * 32;
  const int cj0b = cj0a + 16;

  v8f acc0 = {}; v8f acc1 = {};
  float ssum = 0.f;
  const float* xb = x + ((size_t)b * HWn + (size_t)s * ROWS_PER_SLICE) * Cc;

  auto stage = [&](int buf, int c0) {
    #pragma unroll
    for (int it = 0; it < 4; ++it) {            // 1024 float4 per chunk / 256 thr
      const int f = it * 256 + tid;
      const int row = f >> 4, c4 = (f & 15) * 4;
      ldsLoad16(xb + (size_t)(c0 + row) * Cc + c4, &tile[buf][row * XPAD + c4]);
    }
  };

  const int NCH = ROWS_PER_SLICE / CHUNK;       // 32
  stage(0, 0);
  for (int ch = 0; ch < NCH; ++ch) {
    WAIT_ASYNC0();
    __syncthreads();
    if (ch + 1 < NCH) stage((ch + 1) & 1, (ch + 1) * CHUNK);
    const float* tl = tile[ch & 1];
    // per-channel partial sums (thread owns column tid&63, 16 rows)
    {
      const int colS = tid & 63, rg = (tid >> 6) * 16;
      #pragma unroll
      for (int i = 0; i < 16; ++i) ssum += tl[(rg + i) * XPAD + colS];
    }
    #pragma unroll
    for (int k0 = 0; k0 < CHUNK; k0 += 4) {
      const int kr = k0 + 2 * kh;
      v2f a, b0f, b1f;
      a.x   = tl[kr * XPAD + ci0 + m];          // A[m][k] = tile[k][ci0+m]
      a.y   = tl[(kr + 1) * XPAD + ci0 + m];
      b0f.x = tl[kr * XPAD + cj0a + m];         // B[k][n] = tile[k][cj0+n]
      b0f.y = tl[(kr + 1) * XPAD + cj0a + m];
      b1f.x = tl[kr * XPAD + cj0b + m];
      b1f.y = tl[(kr + 1) * XPAD + cj0b + m];
      acc0 = __builtin_amdgcn_wmma_f32_16x16x4_f32(false, a, false, b0f, (short)0, acc0, false, false);
      acc1 = __builtin_amdgcn_wmma_f32_16x16x4_f32(false, a, false, b1f, (short)0, acc1, false, false);
    }
  }

  float* pg = ws + PGRAM_OFF + (size_t)(b * SLICES + s) * (Cc * Cc);
  #pragma unroll
  for (int r = 0; r < 8; ++r) {
    const int rowc = ci0 + r + 8 * kh;
    pg[rowc * Cc + cj0a + m] = acc0[r];
    pg[rowc * Cc + cj0b + m] = acc1[r];
  }
  red[tid] = ssum;
  __syncthreads();
  if (tid < Cc) {
    float t = red[tid] + red[tid + 64] + red[tid + 128] + red[tid + 192];
    ws[PSUM_OFF + (b * SLICES + s) * Cc + tid] = t;
  }
}

// ============================================================================
// Kernel 2: reduce partials -> mean and covariance. One block per batch.
// ============================================================================
__global__ __launch_bounds__(256) void wct_finalize_cov(float* __restrict__ ws) {
  __shared__ float smean[Cc];
  const int b = blockIdx.x, tid = threadIdx.x;
  if (tid < Cc) {
    float s = 0.f;
    for (int sl = 0; sl < SLICES; ++sl) s += ws[PSUM_OFF + (b * SLICES + sl) * Cc + tid];
    const float mu = s * (1.0f / (float)HWn);
    smean[tid] = mu;
    ws[MEAN_OFF + b * Cc + tid] = mu;
  }
  __syncthreads();
  const float denom = 1.0f / (float)(HWn - 1);
  #pragma unroll
  for (int e = 0; e < 4; ++e) {
    const int idx = (e * 256 + tid) * 4;          // float4-aligned entry index
    const int i = idx >> 6, j0 = idx & 63;
    float4 g = make_float4(0.f, 0.f, 0.f, 0.f);
    for (int sl = 0; sl < SLICES; ++sl) {
      const float4 v = *(const float4*)&ws[PGRAM_OFF + (size_t)(b * SLICES + sl) * (Cc * Cc) + idx];
      g.x += v.x; g.y += v.y; g.z += v.z; g.w += v.w;
    }
    const float mi = smean[i];
    float4 o;
    o.x = (g.x - (float)HWn * mi * smean[j0 + 0]) * denom;
    o.y = (g.y - (float)HWn * mi * smean[j0 + 1]) * denom;
    o.z = (g.z - (float)HWn * mi * smean[j0 + 2]) * denom;
    o.w = (g.w - (float)HWn * mi * smean[j0 + 3]) * denom;
    *(float4*)&ws[COV_OFF + b * Cc * Cc + idx] = o;
  }
}

// ============================================================================
// Kernel 3: coupled Newton-Schulz -> cov^{+1/2} (Y) and cov^{-1/2} (Z).
// ============================================================================
__global__ __launch_bounds__(256) void wct_newton_schulz(float* __restrict__ ws) {
  __shared__ float Ym[Cc * Cc], Zm[Cc * Cc], Tm[Cc * Cc];
  __shared__ float red[256];
  __shared__ float snorm;
  const int b = blockIdx.x, tid = threadIdx.x;
  const float* A = ws + COV_OFF + b * Cc * Cc;

  float ss = 0.f;
  #pragma unroll
  for (int e = 0; e < 16; ++e) {
    const float v = A[e * 256 + tid];
    Ym[e * 256 + tid] = v;
    ss += v * v;
  }
  red[tid] = ss; __syncthreads();
  for (int off = 128; off > 0; off >>= 1) {
    if (tid < off) red[tid] += red[tid + off];
    __syncthreads();
  }
  if (tid == 0) snorm = sqrtf(red[0]);
  __syncthreads();
  const float c = snorm, inv_c = 1.0f / c;
  #pragma unroll
  for (int e = 0; e < 16; ++e) {
    const int idx = e * 256 + tid;
    const int i = idx >> 6, j = idx & 63;
    Ym[idx] *= inv_c;
    Zm[idx] = (i == j) ? 1.0f : 0.0f;
  }
  __syncthreads();

  const int r = tid & 63;
  const int cb = (tid >> 6) * 16;
  for (int it = 0; it < 12; ++it) {
    float acc[16];
    #pragma unroll
    for (int j = 0; j < 16; ++j) acc[j] = 0.f;
    for (int k = 0; k < Cc; ++k) {
      const float zk = Zm[r * Cc + k];
      #pragma unroll
      for (int j = 0; j < 16; ++j) acc[j] += zk * Ym[k * Cc + cb + j];
    }
    #pragma unroll
    for (int j = 0; j < 16; ++j) {
      const int col = cb + j;
      Tm[r * Cc + col] = ((r == col) ? 1.5f : 0.0f) - 0.5f * acc[j];
    }
    __syncthreads();
    float ya[16], za[16];
    #pragma unroll
    for (int j = 0; j < 16; ++j) { ya[j] = 0.f; za[j] = 0.f; }
    for (int k = 0; k < Cc; ++k) {
      const float yk = Ym[r * Cc + k];
      const float tk = Tm[r * Cc + k];
      #pragma unroll
      for (int j = 0; j < 16; ++j) {
        ya[j] += yk * Tm[k * Cc + cb + j];
        za[j] += tk * Zm[k * Cc + cb + j];
      }
    }
    __syncthreads();
    #pragma unroll
    for (int j = 0; j < 16; ++j) { Ym[r * Cc + cb + j] = ya[j]; Zm[r * Cc + cb + j] = za[j]; }
    __syncthreads();
  }
  const float sc = sqrtf(c), isc = 1.0f / sc;
  #pragma unroll
  for (int e = 0; e < 16; ++e) {
    const int idx = e * 256 + tid;
    ws[YSQ_OFF + b * Cc * Cc + idx] = Ym[idx] * sc;
    ws[ZIN_OFF + b * Cc * Cc + idx] = Zm[idx] * isc;
  }
}

// ============================================================================
// Kernel 4: Mt[b] = Z[b] @ Y[perm[b]] ; bias[b] = mean[perm[b]] - mean[b]@Mt.
// ============================================================================
__global__ __launch_bounds__(256) void wct_compose(float* __restrict__ ws,
                                                   const int* __restrict__ perm) {
  __shared__ float Zs[Cc * Cc], Ys[Cc * Cc], Ms[Cc * Cc];
  const int b = blockIdx.x, tid = threadIdx.x;
  const int p = perm[b];
  #pragma unroll
  for (int e = 0; e < 16; ++e) {
    const int idx = e * 256 + tid;
    Zs[idx] = ws[ZIN_OFF + b * Cc * Cc + idx];
    Ys[idx] = ws[YSQ_OFF + p * Cc * Cc + idx];
  }
  __syncthreads();
  const int r = tid & 63, cb = (tid >> 6) * 16;
  float acc[16];
  #pragma unroll
  for (int j = 0; j < 16; ++j) acc[j] = 0.f;
  for (int k = 0; k < Cc; ++k) {
    const float zk = Zs[r * Cc + k];
    #pragma unroll
    for (int j = 0; j < 16; ++j) acc[j] += zk * Ys[k * Cc + cb + j];
  }
  #pragma unroll
  for (int j = 0; j < 16; ++j) {
    Ms[r * Cc + cb + j] = acc[j];
    ws[MT_OFF + b * Cc * Cc + r * Cc + cb + j] = acc[j];
  }
  __syncthreads();
  if (tid < Cc) {
    const float mu_s = ws[MEAN_OFF + p * Cc + tid];
    float bsum = 0.f;
    for (int i = 0; i < Cc; ++i) bsum += ws[MEAN_OFF + b * Cc + i] * Ms[i * Cc + tid];
    ws[BIAS_OFF + b * Cc + tid] = mu_s - bsum;
  }
}

// ============================================================================
// Kernel 5: apply. out_row = alpha*x_row + (1-alpha)*(x_row @ Mt + bias).
// Async-stage Mt + 128 rows; A-fragment shared across the 4 N-tiles; blended
// results written back into the LDS tile, then coalesced 128-bit stores.
// ============================================================================
__global__ __launch_bounds__(256) void wct_apply(const float* __restrict__ x,
                                                 const float* __restrict__ alpha,
                                                 const float* __restrict__ ws,
                                                 float* __restrict__ out) {
  __shared__ float Msh[Cc * XPAD];
  __shared__ float Xs[128 * XPAD];
  __shared__ float sbias[Cc];
  const int ROWS = 128;
  const int b  = blockIdx.x / (HWn / ROWS);
  const int rb = blockIdx.x % (HWn / ROWS);
  const int tid = threadIdx.x, lane = tid & 31, wave = tid >> 5;

  const float* mtb = ws + MT_OFF + b * Cc * Cc;
  const float* xb  = x + ((size_t)b * HWn + (size_t)rb * ROWS) * Cc;
  float* ob = out + ((size_t)b * HWn + (size_t)rb * ROWS) * Cc;

  #pragma unroll
  for (int it = 0; it < 4; ++it) {              // Mt: 1024 float4
    const int f = it * 256 + tid;
    const int row = f >> 4, c4 = (f & 15) * 4;
    ldsLoad16(mtb + row * Cc + c4, &Msh[row * XPAD + c4]);
  }
  #pragma unroll
  for (int it = 0; it < 8; ++it) {              // X: 2048 float4
    const int f = it * 256 + tid;
    const int row = f >> 4, c4 = (f & 15) * 4;
    ldsLoad16(xb + (size_t)row * Cc + c4, &Xs[row * XPAD + c4]);
  }
  if (tid < Cc) sbias[tid] = ws[BIAS_OFF + b * Cc + tid];
  const float al = alpha[b];
  const float om = 1.0f - al;
  WAIT_ASYNC0();
  __syncthreads();

  const int m = lane & 15, kh = lane >> 4;
  const int r0 = wave * 16;
  v8f acc[4] = {v8f{}, v8f{}, v8f{}, v8f{}};
  #pragma unroll
  for (int k0 = 0; k0 < Cc; k0 += 4) {
    const int kr = k0 + 2 * kh;
    v2f a;
    a.x = Xs[(r0 + m) * XPAD + kr];
    a.y = Xs[(r0 + m) * XPAD + kr + 1];
    #pragma unroll
    for (int nt = 0; nt < 4; ++nt) {
      v2f bb;
      bb.x = Msh[kr * XPAD + nt * 16 + m];
      bb.y = Msh[(kr + 1) * XPAD + nt * 16 + m];
      acc[nt] = __builtin_amdgcn_wmma_f32_16x16x4_f32(false, a, false, bb, (short)0, acc[nt], false, false);
    }
  }
  // blend + write back into the LDS tile (each (row,col) owned by one lane)
  #pragma unroll
  for (int nt = 0; nt < 4; ++nt) {
    #pragma unroll
    for (int r = 0; r < 8; ++r) {
      const int row = r0 + r + 8 * kh;
      const int col = nt * 16 + m;
      const float xv = Xs[row * XPAD + col];
      Xs[row * XPAD + col] = al * xv + om * (acc[nt][r] + sbias[col]);
    }
  }
  __syncthreads();
  // coalesced 128-bit stores (async LDS->global when available)
  #pragma unroll
  for (int e = 0; e < 8; ++e) {
    const int f = e * 256 + tid;
    const int row = f >> 4, c4 = (f & 15) * 4;
    ldsStore16(ob + (size_t)row * Cc + c4, &Xs[row * XPAD + c4]);
  }
#ifdef HAVE_ASYNC_STORE
  WAIT_ASYNC0();
#endif
}

extern "C" void kernel_launch(void* const* d_in, const int* in_sizes, int n_in,
                              void* d_out, int out_size, void* d_ws, size_t ws_size,
                              hipStream_t stream) {
  (void)in_sizes; (void)n_in; (void)out_size; (void)ws_size;
  const float* x     = (const float*)d_in[0];
  const int*   perm  = (const int*)d_in[1];
  const float* alpha = (const float*)d_in[2];
  float* out = (float*)d_out;
  float* ws  = (float*)d_ws;

  wct_gram_partial<<<Bn * SLICES, 256, 0, stream>>>(x, ws);
  wct_finalize_cov<<<Bn, 256, 0, stream>>>(ws);
  wct_newton_schulz<<<Bn, 256, 0, stream>>>(ws);
  wct_compose<<<Bn, 256, 0, stream>>>(ws, perm);
  wct_apply<<<Bn * (HWn / 128), 256, 0, stream>>>(x, alpha, ws, out);
}